// lrRNN_6253472383742
// MI455X (gfx1250) — compile-verified
//
#include <hip/hip_runtime.h>
#include <math.h>
#include <stdint.h>

// Problem constants (from reference)
constexpr int   BB = 64, TT = 1024, HH = 1024, II = 8, OO = 8, RR = 4;
constexpr float K_NOISE = 0.05f, K_ALPHA = 0.2f;

constexpr int THREADS = 256;        // 8 wave32 waves
constexpr int NWAVES  = THREADS / 32;
constexpr int HPT     = HH / THREADS;   // 4 hidden units per thread
constexpr int CHUNK   = 16;             // timesteps per TDM chunk (64 KB)
constexpr int NCH     = TT / CHUNK;

typedef float    v2f __attribute__((ext_vector_type(2)));
typedef float    v8f __attribute__((ext_vector_type(8)));
typedef unsigned v4u __attribute__((ext_vector_type(4)));
typedef int      v8i __attribute__((ext_vector_type(8)));
typedef int      v4i __attribute__((ext_vector_type(4)));

__device__ __forceinline__ float fast_tanh(float v) {
#if __has_builtin(__builtin_amdgcn_tanhf)
    return __builtin_amdgcn_tanhf(v);
#elif __has_builtin(__builtin_amdgcn_tanh_f32)
    return __builtin_amdgcn_tanh_f32(v);
#else
    return tanhf(v);
#endif
}

__device__ __forceinline__ float wred(float x) {
#pragma unroll
    for (int s = 16; s > 0; s >>= 1) x += __shfl_xor(x, s, 32);
    return x;
}

// Issue one TDM 1-D copy: CHUNK*HH f32 elements, global -> LDS (async, TENSORcnt).
// D# packed per CDNA5 ISA §8 (group0: count/lds/global/type, group1: dims/tile).
// clang-23 toolchain: 6-arg builtin (g0 u32x4, g1 i32x8, g2 i32x4, g3 i32x4, i32x8, cpol).
__device__ __forceinline__ void tdm_load_chunk(const float* gsrc, uint32_t lds_off) {
    uint64_t ga = (uint64_t)(uintptr_t)gsrc;
    v4u g0;
    g0[0] = 1u;                                                   // count=1, user mode
    g0[1] = lds_off;                                              // lds_addr (bytes)
    g0[2] = (uint32_t)ga;                                         // global_addr[31:0]
    g0[3] = (uint32_t)((ga >> 32) & 0x01FFFFFFu) | (2u << 30);    // addr[56:32] | type=2
    constexpr unsigned ELEMS = (unsigned)(CHUNK * HH);            // 16384 (fits tile_dim0 u16)
    v8i g1;
    g1[0] = (int)(2u << 16);                  // wg_mask=0, data_size=2 (4 bytes)
    g1[1] = (int)((ELEMS & 0xFFFFu) << 16);   // tensor_dim0[15:0]
    g1[2] = (int)((ELEMS >> 16) | (1u << 16)); // tensor_dim0[31:16], tensor_dim1=1 (lo)
    g1[3] = (int)((ELEMS & 0xFFFFu) << 16);   // tensor_dim1 hi=0, tile_dim0=ELEMS
    g1[4] = 1;                                // tile_dim1=1, tile_dim2=0
    g1[5] = (int)ELEMS;                       // tensor_dim0_stride[31:0]
    g1[6] = 0;
    g1[7] = 0;
    v4i z4 = {0, 0, 0, 0};
    v8i z8 = {0, 0, 0, 0, 0, 0, 0, 0};
    __builtin_amdgcn_tensor_load_to_lds(g0, g1, z4, z4, z8, 0);
}

__global__ __launch_bounds__(THREADS, 1)
void lrrnn_scan_kernel(const float* __restrict__ x,  const float* __restrict__ noise,
                       const float* __restrict__ wi, const float* __restrict__ si,
                       const float* __restrict__ m,  const float* __restrict__ n,
                       const float* __restrict__ bb, const float* __restrict__ wo,
                       const float* __restrict__ so, const float* __restrict__ h0,
                       float* __restrict__ out)
{
    __shared__ __align__(16) float NZ[2][CHUNK * HH];   // 128 KB noise double buffer (TDM dest)
    __shared__ __align__(16) float XS[TT * II];         // 32 KB: x[b] staged once
    __shared__ __align__(16) float PP[2][NWAVES * 16];  // per-step partials (double buffered)
    __shared__ __align__(16) float PRE[NWAVES * RR];    // initial v reduction

    const int tid   = threadIdx.x;
    const int wid   = tid >> 5;
    const int lane  = tid & 31;
    const int b     = blockIdx.x;
    const int hbase = tid * HPT;

    // ---- stage x[b] (T x 8 f32) into LDS, coalesced ----
    {
        const float4* src = (const float4*)(x + (size_t)b * TT * II);
        float4* dst = (float4*)XS;
#pragma unroll
        for (int k = 0; k < (TT * II / 4) / THREADS; ++k)
            dst[tid + k * THREADS] = src[tid + k * THREADS];
    }
    if (tid < 2 * NWAVES * 16) ((float*)PP)[tid] = 0.0f;   // rows 12..15 stay 0 forever

    // ---- per-thread constants -> VGPRs ----
    float sv[II], ov[OO];
    {
        float4 s0 = *(const float4*)(si), s1 = *(const float4*)(si + 4);
        sv[0]=s0.x; sv[1]=s0.y; sv[2]=s0.z; sv[3]=s0.w; sv[4]=s1.x; sv[5]=s1.y; sv[6]=s1.z; sv[7]=s1.w;
        float4 o0 = *(const float4*)(so), o1 = *(const float4*)(so + 4);
        ov[0]=o0.x; ov[1]=o0.y; ov[2]=o0.z; ov[3]=o0.w; ov[4]=o1.x; ov[5]=o1.y; ov[6]=o1.z; ov[7]=o1.w;
    }
    float wi4[II][HPT];
#pragma unroll
    for (int i = 0; i < II; ++i) {
        float4 w = *(const float4*)(wi + (size_t)i * HH + hbase);
        wi4[i][0] = w.x * sv[i]; wi4[i][1] = w.y * sv[i];
        wi4[i][2] = w.z * sv[i]; wi4[i][3] = w.w * sv[i];
    }
    float n4[HPT][RR], m4[HPT][RR], wo4[HPT][OO], b4[HPT], h[HPT], r[HPT];
    constexpr float invH = 1.0f / (float)HH;
#pragma unroll
    for (int j = 0; j < HPT; ++j) {
        float4 nn = *(const float4*)(n + (size_t)(hbase + j) * RR);
        n4[j][0]=nn.x; n4[j][1]=nn.y; n4[j][2]=nn.z; n4[j][3]=nn.w;
        float4 mm = *(const float4*)(m + (size_t)(hbase + j) * RR);
        m4[j][0]=mm.x*invH; m4[j][1]=mm.y*invH; m4[j][2]=mm.z*invH; m4[j][3]=mm.w*invH;
        float4 w0 = *(const float4*)(wo + (size_t)(hbase + j) * OO);
        float4 w1 = *(const float4*)(wo + (size_t)(hbase + j) * OO + 4);
        wo4[j][0]=w0.x*ov[0]; wo4[j][1]=w0.y*ov[1]; wo4[j][2]=w0.z*ov[2]; wo4[j][3]=w0.w*ov[3];
        wo4[j][4]=w1.x*ov[4]; wo4[j][5]=w1.y*ov[5]; wo4[j][6]=w1.z*ov[6]; wo4[j][7]=w1.w*ov[7];
    }
    {
        float4 bv = *(const float4*)(bb + hbase);
        b4[0]=bv.x; b4[1]=bv.y; b4[2]=bv.z; b4[3]=bv.w;
        float4 hv = *(const float4*)(h0 + hbase);
        h[0]=hv.x; h[1]=hv.y; h[2]=hv.z; h[3]=hv.w;
#pragma unroll
        for (int j = 0; j < HPT; ++j) r[j] = fast_tanh(h[j]);   // r_init = tanh(h0), no bias
    }

    // ---- kick off TDM prefetch of first two noise chunks (wave 0 issues) ----
    if (wid == 0) {
        tdm_load_chunk(noise + ((size_t)b * TT + 0 * CHUNK) * HH, (uint32_t)(uintptr_t)&NZ[0][0]);
        tdm_load_chunk(noise + ((size_t)b * TT + 1 * CHUNK) * HH, (uint32_t)(uintptr_t)&NZ[1][0]);
    }

    // ---- initial v = r_init @ n (4 scalars) ----
    float vv[RR];
    {
        float pv[RR] = {0.f, 0.f, 0.f, 0.f};
#pragma unroll
        for (int j = 0; j < HPT; ++j)
#pragma unroll
            for (int q = 0; q < RR; ++q) pv[q] = fmaf(r[j], n4[j][q], pv[q]);
#pragma unroll
        for (int q = 0; q < RR; ++q) pv[q] = wred(pv[q]);
        if (lane == 0) *(float4*)(&PRE[wid * RR]) = make_float4(pv[0], pv[1], pv[2], pv[3]);
    }
    __syncthreads();
    {
        float a0=0.f, a1=0.f, a2=0.f, a3=0.f;
#pragma unroll
        for (int w = 0; w < NWAVES; ++w) {
            float4 p = *(float4*)(&PRE[w * RR]);
            a0 += p.x; a1 += p.y; a2 += p.z; a3 += p.w;
        }
        vv[0]=a0; vv[1]=a1; vv[2]=a2; vv[3]=a3;
    }

    float* outBase = out + (size_t)b * TT * OO;

    // ================= sequential scan =================
    for (int c = 0; c < NCH; ++c) {
        if (wid == 0) {
            if (c + 1 < NCH) __builtin_amdgcn_s_wait_tensorcnt(1);  // oldest (chunk c) done
            else             __builtin_amdgcn_s_wait_tensorcnt(0);
        }
        __syncthreads();                       // chunk c resident in NZ[c&1]
        const float* nzbuf = &NZ[c & 1][0];

        for (int tt = 0; tt < CHUNK; ++tt) {
            const int t = c * CHUNK + tt;
            // per-step inputs
            float4 nz = *(const float4*)(nzbuf + tt * HH + hbase);       // ds_load_b128
            float4 xa = *(const float4*)(&XS[t * II]);                   // broadcast
            float4 xb = *(const float4*)(&XS[t * II + 4]);
            const float xr[II]  = {xa.x, xa.y, xa.z, xa.w, xb.x, xb.y, xb.z, xb.w};
            const float nzr[HPT] = {nz.x, nz.y, nz.z, nz.w};

            float pv[RR] = {0.f, 0.f, 0.f, 0.f};
            float py[OO] = {0.f, 0.f, 0.f, 0.f, 0.f, 0.f, 0.f, 0.f};
#pragma unroll
            for (int j = 0; j < HPT; ++j) {
                float u = 0.f;
#pragma unroll
                for (int i = 0; i < II; ++i) u = fmaf(wi4[i][j], xr[i], u);
                float rec = 0.f;
#pragma unroll
                for (int q = 0; q < RR; ++q) rec = fmaf(m4[j][q], vv[q], rec);
                float hj = h[j];
                hj = hj + K_NOISE * nzr[j] + K_ALPHA * (rec + u - hj);
                h[j] = hj;
                float rj = fast_tanh(hj + b4[j]);
                r[j] = rj;
#pragma unroll
                for (int q = 0; q < RR; ++q) pv[q] = fmaf(rj, n4[j][q], pv[q]);
#pragma unroll
                for (int o = 0; o < OO; ++o) py[o] = fmaf(hj, wo4[j][o], py[o]);
            }

            // intra-wave xor-tree reduction of the 12 scalars
#pragma unroll
            for (int q = 0; q < RR; ++q) pv[q] = wred(pv[q]);
#pragma unroll
            for (int o = 0; o < OO; ++o) py[o] = wred(py[o]);

            float* pbuf = &PP[t & 1][0];
            if (lane == 0) {
                *(float4*)(pbuf + wid * 16 + 0) = make_float4(pv[0], pv[1], pv[2], pv[3]);
                *(float4*)(pbuf + wid * 16 + 4) = make_float4(py[0], py[1], py[2], py[3]);
                *(float4*)(pbuf + wid * 16 + 8) = make_float4(py[4], py[5], py[6], py[7]);
            }
            __syncthreads();   // the ONLY per-step barrier

            // cross-wave reduction as a rank-1 WMMA: C[m,:] = sum_k A[m,k] * 1
            // A (16x4 f32 frag): lanes0-15 M=lane {K,K+1}, lanes16-31 M=lane-16 {K+2,K+3}
            const int mm = lane & 15;
            const int kb = (lane >> 4) << 1;   // 0 or 2
            v2f ones; ones[0] = 1.0f; ones[1] = 1.0f;
            v2f a0, a1;
            a0[0] = pbuf[(kb + 0) * 16 + mm];
            a0[1] = pbuf[(kb + 1) * 16 + mm];
            a1[0] = pbuf[(kb + 4) * 16 + mm];
            a1[1] = pbuf[(kb + 5) * 16 + mm];
            v8f C = {};
            C = __builtin_amdgcn_wmma_f32_16x16x4_f32(false, a0, false, ones, (short)0, C, false, false);
            C = __builtin_amdgcn_wmma_f32_16x16x4_f32(false, a1, false, ones, (short)0, C, false, false);

            // v for step t+1: rows 0..3 (lanes<16 hold M=0..7); mirror into upper half
#pragma unroll
            for (int q = 0; q < RR; ++q) {
                float mir = __shfl_xor(C[q], 16, 32);
                vv[q] = (lane < 16) ? C[q] : mir;
            }

            // output y_t: rows 4..7 -> lanes<16 C[4..7]; rows 8..11 -> lanes>=16 C[0..3]
            if (wid == 0) {
                float* op = outBase + (size_t)t * OO;
                if (lane == 0)       *(float4*)(op)     = make_float4(C[4], C[5], C[6], C[7]);
                else if (lane == 16) *(float4*)(op + 4) = make_float4(C[0], C[1], C[2], C[3]);
            }
        }

        // prefetch chunk c+2 into the buffer just drained (reads all fenced by last barrier)
        if (wid == 0 && c + 2 < NCH)
            tdm_load_chunk(noise + ((size_t)b * TT + (size_t)(c + 2) * CHUNK) * HH,
                           (uint32_t)(uintptr_t)&NZ[c & 1][0]);
    }
}

extern "C" void kernel_launch(void* const* d_in, const int* in_sizes, int n_in,
                              void* d_out, int out_size, void* d_ws, size_t ws_size,
                              hipStream_t stream) {
    const float* x     = (const float*)d_in[0];
    const float* noise = (const float*)d_in[1];
    const float* wi    = (const float*)d_in[2];
    const float* si    = (const float*)d_in[3];
    const float* m     = (const float*)d_in[4];
    const float* n     = (const float*)d_in[5];
    const float* bb    = (const float*)d_in[6];
    const float* wo    = (const float*)d_in[7];
    const float* so    = (const float*)d_in[8];
    const float* h0    = (const float*)d_in[9];
    float* out = (float*)d_out;
    (void)in_sizes; (void)n_in; (void)out_size; (void)d_ws; (void)ws_size;
    hipLaunchKernelGGL(lrrnn_scan_kernel, dim3(BB), dim3(THREADS), 0, stream,
                       x, noise, wi, si, m, n, bb, wo, so, h0, out);
}